// FlexPool_45535243273089
// MI455X (gfx1250) — compile-verified
//
#include <hip/hip_runtime.h>
#include <math.h>
#include <stdint.h>

// ---------------------------------------------------------------------------
// BFP quantize over channel blocks of an NCHW fp32 tensor (mantissa_bits=3).
//   per (n,h,w): absmax over C, E=floor(log2(absmax)), ulp=2^(E-2),
//   q = clip(rint(x/ulp), -7, 7) * ulp    (0 if absmax==0)
//
// Bandwidth-bound: 205.5 MB in + 205.5 MB out -> ~17.6 us floor @ 23.3 TB/s.
// Strategy: stage a (C=256) x (TILE=128 hw) tile = 128 KB in LDS via gfx1250
// async global->LDS copies (one HBM read per element), reduce + quantize out
// of LDS, write once with NT stores. 128 KB LDS -> 2 workgroups per WGP.
// Full tiles (24 of 25 per image) take an unpredicated straight-line async
// path (EXEC all-1s); only the tail tile uses a lane-guarded fallback.
// ---------------------------------------------------------------------------

#define C_DIM   256
#define HW_DIM  3136          // 56*56
#define TILE    128           // hw positions per workgroup
#define THREADS 128           // 4 waves (wave32)
#define NWAVES  (THREADS / 32)
#define MANT_MAX 7.0f         // 2^3 - 1

#if defined(__gfx1250__) && __has_builtin(__builtin_amdgcn_global_load_async_to_lds_b128)
#define USE_ASYNC_LDS 1
#else
#define USE_ASYNC_LDS 0
#endif

// Builtin signature (from hipcc diagnostic): param 1 is
//   '__attribute__((__vector_size__(4 * sizeof(int)))) int __device__ *'
// i.e. AS(1) pointer to a GCC-style int4 vector; LDS side is the AS(3) twin.
typedef int v4i_t __attribute__((vector_size(4 * sizeof(int))));
typedef __attribute__((address_space(1))) v4i_t glb_v4i_t;
typedef __attribute__((address_space(3))) v4i_t lds_v4i_t;

__global__ __launch_bounds__(THREADS) void FlexPool_bfp_quant_kernel(
    const float* __restrict__ in, float* __restrict__ out) {
  extern __shared__ float smem[];  // [C_DIM][TILE] floats = 128 KB

  const int n      = blockIdx.y;
  const int hw0    = blockIdx.x * TILE;
  const int t      = threadIdx.x;
  const int wave   = t >> 5;
  const int lane   = t & 31;
  const int nvalid = min(TILE, HW_DIM - hw0);   // last tile holds 64 positions

  // ---- Stage tile into LDS ------------------------------------------------
  // Wave w copies channel rows c = w, w+4, ...; each lane moves 16 B so a
  // single b128 async op per wave stages one full 128-float channel row.
  // Block-uniform split: full tiles run unpredicated (EXEC all-1s).
  if (hw0 + TILE <= HW_DIM) {                    // uniform: full tile
    const int hw = hw0 + (lane << 2);            // 4 floats per lane
#if USE_ASYNC_LDS
#pragma unroll 4
    for (int c = wave; c < C_DIM; c += NWAVES) {
      const float* gsrc = in + (size_t)(n * C_DIM + c) * HW_DIM + hw;
      float*       lptr = smem + c * TILE + (lane << 2);
      __builtin_amdgcn_global_load_async_to_lds_b128(
          (glb_v4i_t*)(uintptr_t)gsrc,
          (lds_v4i_t*)(uintptr_t)lptr,           // generic LDS addr: offset in low 32b
          0 /*imm offset*/, 0 /*cpol*/);
    }
#if __has_builtin(__builtin_amdgcn_s_wait_asynccnt)
    __builtin_amdgcn_s_wait_asynccnt(0);
#else
    asm volatile("s_wait_asynccnt 0x0" ::: "memory");
#endif
#else
#pragma unroll 4
    for (int c = wave; c < C_DIM; c += NWAVES) {
      const float4 v = *(const float4*)(in + (size_t)(n * C_DIM + c) * HW_DIM + hw);
      *(float4*)(smem + c * TILE + (lane << 2)) = v;
    }
#endif
  } else {                                       // uniform: tail tile (1 per image)
    const int hw = hw0 + (lane << 2);
    if (hw < HW_DIM) {                           // lane-granular guard, plain copies
      for (int c = wave; c < C_DIM; c += NWAVES) {
        const float4 v = *(const float4*)(in + (size_t)(n * C_DIM + c) * HW_DIM + hw);
        *(float4*)(smem + c * TILE + (lane << 2)) = v;
      }
    }
  }
  __syncthreads();

  if (t < nvalid) {                              // tail tile: skip idle threads
    // ---- Pass 1: absmax over the channel column (conflict-free LDS reads) -
    float am0 = 0.0f, am1 = 0.0f, am2 = 0.0f, am3 = 0.0f;
#pragma unroll 4
    for (int c = 0; c < C_DIM; c += 4) {
      am0 = fmaxf(am0, fabsf(smem[(c + 0) * TILE + t]));
      am1 = fmaxf(am1, fabsf(smem[(c + 1) * TILE + t]));
      am2 = fmaxf(am2, fabsf(smem[(c + 2) * TILE + t]));
      am3 = fmaxf(am3, fabsf(smem[(c + 3) * TILE + t]));
    }
    const float am = fmaxf(fmaxf(am0, am1), fmaxf(am2, am3));

    // ---- Shared exponent: E = floor(log2(am)), exact via frexp ------------
    int e = 1;                                   // am==0 -> safe=1 -> E=0
    if (am > 0.0f) (void)frexpf(am, &e);         // am = m*2^e, m in [0.5,1)
    const int   E   = e - 1;
    const float ulp = ldexpf(1.0f, E - 2);       // 2^(E+1-mant_bits)
    const float inv = ldexpf(1.0f, 2 - E);       // exact reciprocal (pow2)

    // ---- Pass 2: quantize + write-once (NT stores: output never re-read) --
    float* op = out + (size_t)n * C_DIM * HW_DIM + hw0 + t;
#pragma unroll 4
    for (int c = 0; c < C_DIM; ++c) {
      const float x = smem[c * TILE + t];
      float r = rintf(x * inv);                  // RTE, matches jnp.round
      r = fminf(MANT_MAX, fmaxf(-MANT_MAX, r));
      __builtin_nontemporal_store(r * ulp, op + (size_t)c * HW_DIM);
    }
  }
}

extern "C" void kernel_launch(void* const* d_in, const int* in_sizes, int n_in,
                              void* d_out, int out_size, void* d_ws, size_t ws_size,
                              hipStream_t stream) {
  (void)n_in; (void)out_size; (void)d_ws; (void)ws_size;
  const float* in  = (const float*)d_in[0];
  float*       out = (float*)d_out;

  const int N = in_sizes[0] / (C_DIM * HW_DIM);  // 64
  dim3 grid((HW_DIM + TILE - 1) / TILE, N, 1);   // (25, 64)
  dim3 block(THREADS, 1, 1);
  const size_t smem_bytes = (size_t)C_DIM * TILE * sizeof(float);  // 128 KB

  FlexPool_bfp_quant_kernel<<<grid, block, smem_bytes, stream>>>(in, out);
}